// GinConvLayer_36696200577450
// MI455X (gfx1250) — compile-verified
//
#include <hip/hip_runtime.h>

#define N_NODES 50000
#define N_EDGES 800000
#define BATCH   4
#define C       64
#define M_ROWS  (BATCH * N_NODES)   // 200000 rows in the flattened (B*N, C) view
#define M_TILES (M_ROWS / 16)       // 12500 (exact)

typedef __attribute__((ext_vector_type(2))) float v2f;
typedef __attribute__((ext_vector_type(8))) float v8f;

// ---------------------------------------------------------------------------
// Zero-fill workspace (graph-capture safe; no hipMemsetAsync dependence)
// ---------------------------------------------------------------------------
__global__ void gin_zero_kernel(float4* __restrict__ p, int n4) {
    int i = blockIdx.x * blockDim.x + threadIdx.x;
    if (i < n4) p[i] = make_float4(0.f, 0.f, 0.f, 0.f);
}

// ---------------------------------------------------------------------------
// Scatter-add: agg[b, dst[e], c] += h[b, src[e], c]
// One thread per (edge, channel); 64 consecutive threads share an edge ->
// coalesced gather + coalesced atomic burst. agg (51.2MB) is L2-resident.
// ---------------------------------------------------------------------------
__global__ void gin_scatter_kernel(const float* __restrict__ h,
                                   const long long* __restrict__ src,
                                   const long long* __restrict__ dst,
                                   float* __restrict__ agg) {
    long long t = (long long)blockIdx.x * blockDim.x + threadIdx.x;
    int e = (int)(t >> 6);
    int c = (int)(t & 63);
    if (e >= N_EDGES) return;
    int s = (int)src[e];
    int d = (int)dst[e];
    const float* hp = h   + (size_t)s * C + c;
    float*       ap = agg + (size_t)d * C + c;
#pragma unroll
    for (int b = 0; b < BATCH; ++b) {
        // global_atomic_add_f32 (fast L2-side fp32 atomic)
        unsafeAtomicAdd(ap + (size_t)b * N_NODES * C,
                        hp[(size_t)b * N_NODES * C]);
    }
}

// ---------------------------------------------------------------------------
// Fused GIN update GEMM: out = (hin + agg) @ W^T + bias
// M = 200000, K = 64, Nout = 64.  One wave32 per 16-row stripe; it produces
// all 4 16x16 output tiles using V_WMMA_F32_16X16X4_F32 (full fp32 precision).
//
// f32 16x4 A-fragment layout (per ISA): lane<16 -> row M=lane, v0=K0,v1=K1;
// lane>=16 -> row M=lane-16, v0=K2,v1=K3.  So each lane loads a float2 at
// column kc + 2*(lane>>4).  B (4x16 = W^T chunk) is symmetric.
// C/D layout: VGPR i, lanes 0-15 -> M=i, lanes 16-31 -> M=8+i, N = lane&15.
// ---------------------------------------------------------------------------
__global__ void gin_gemm_kernel(const float* __restrict__ hin,
                                const float* __restrict__ agg,
                                const float* __restrict__ W,    // (64 out, 64 in)
                                const float* __restrict__ bias, // (64)
                                float* __restrict__ out) {
    int wave = blockIdx.x * (blockDim.x >> 5) + (threadIdx.x >> 5);
    if (wave >= M_TILES) return;              // wave-uniform: EXEC stays all-1s
    int lane = threadIdx.x & 31;
    int half = lane >> 4;                     // 0 or 1
    int l15  = lane & 15;

    v8f c0 = {}, c1 = {}, c2 = {}, c3 = {};

#if defined(__has_builtin) && __has_builtin(__builtin_amdgcn_wmma_f32_16x16x4_f32)
    size_t arow = (size_t)wave * 16 + l15;    // this lane's A row
#pragma unroll
    for (int kc = 0; kc < C; kc += 4) {
        int col = kc + 2 * half;
        const float* ap = hin + arow * C + col;
        const float* gp = agg + arow * C + col;
        v2f a;
        a.x = ap[0] + gp[0];
        a.y = ap[1] + gp[1];

        const float* wp = W + (size_t)l15 * C + col;   // W[n][k] = B[k][n]
        v2f b0 = { wp[0 * 16 * C + 0], wp[0 * 16 * C + 1] };
        v2f b1 = { wp[1 * 16 * C + 0], wp[1 * 16 * C + 1] };
        v2f b2 = { wp[2 * 16 * C + 0], wp[2 * 16 * C + 1] };
        v2f b3 = { wp[3 * 16 * C + 0], wp[3 * 16 * C + 1] };

        c0 = __builtin_amdgcn_wmma_f32_16x16x4_f32(false, a, false, b0, (short)0, c0, false, false);
        c1 = __builtin_amdgcn_wmma_f32_16x16x4_f32(false, a, false, b1, (short)0, c1, false, false);
        c2 = __builtin_amdgcn_wmma_f32_16x16x4_f32(false, a, false, b2, (short)0, c2, false, false);
        c3 = __builtin_amdgcn_wmma_f32_16x16x4_f32(false, a, false, b3, (short)0, c3, false, false);
    }
#else
    // Scalar fallback with identical register semantics (correct, no wmma).
    for (int i = 0; i < 8; ++i) {
        size_t r = (size_t)wave * 16 + 8 * half + i;
        float a0 = 0.f, a1 = 0.f, a2 = 0.f, a3 = 0.f;
        for (int k = 0; k < C; ++k) {
            float hv = hin[r * C + k] + agg[r * C + k];
            a0 += hv * W[(size_t)(0 * 16 + l15) * C + k];
            a1 += hv * W[(size_t)(1 * 16 + l15) * C + k];
            a2 += hv * W[(size_t)(2 * 16 + l15) * C + k];
            a3 += hv * W[(size_t)(3 * 16 + l15) * C + k];
        }
        c0[i] = a0; c1[i] = a1; c2[i] = a2; c3[i] = a3;
    }
#endif

    // Store with fused bias.  Lane's N is fixed per tile; row = base + 8*half + i.
    size_t row0 = (size_t)wave * 16 + 8 * half;
    float bs0 = bias[ 0 + l15];
    float bs1 = bias[16 + l15];
    float bs2 = bias[32 + l15];
    float bs3 = bias[48 + l15];
#pragma unroll
    for (int i = 0; i < 8; ++i) {
        float* op = out + (row0 + i) * C + l15;
        op[ 0] = c0[i] + bs0;
        op[16] = c1[i] + bs1;
        op[32] = c2[i] + bs2;
        op[48] = c3[i] + bs3;
    }
}

// ---------------------------------------------------------------------------
// Launcher: zero -> scatter -> gemm, twice.  Layer-2 GEMM is safely in-place
// on d_out (each wave reads only its own 16 rows before writing them).
// Workspace use: agg buffer only (B*N*C floats = 51.2 MB).
// ---------------------------------------------------------------------------
extern "C" void kernel_launch(void* const* d_in, const int* in_sizes, int n_in,
                              void* d_out, int out_size, void* d_ws, size_t ws_size,
                              hipStream_t stream) {
    const float*     x   = (const float*)d_in[0];
    const long long* ei  = (const long long*)d_in[1];   // int64 (2, E)
    const float*     W1  = (const float*)d_in[2];
    const float*     b1  = (const float*)d_in[3];
    const float*     W2  = (const float*)d_in[4];
    const float*     b2  = (const float*)d_in[5];
    float* out = (float*)d_out;
    float* agg = (float*)d_ws;

    const long long* src = ei;             // edge_index[0]
    const long long* dst = ei + N_EDGES;   // edge_index[1]

    const int n4 = (M_ROWS * C) / 4;
    dim3 zblk(256), zgrd((n4 + 255) / 256);

    const long long sthreads = (long long)N_EDGES * 64;   // 51.2M
    dim3 sblk(256), sgrd((unsigned)((sthreads + 255) / 256));

    dim3 gblk(256), ggrd((M_TILES + 7) / 8);              // 8 waves / block

    // Layer 1: agg1 = scatter(x); h1 = (x + agg1) @ W1^T + b1   -> d_out
    gin_zero_kernel<<<zgrd, zblk, 0, stream>>>((float4*)agg, n4);
    gin_scatter_kernel<<<sgrd, sblk, 0, stream>>>(x, src, dst, agg);
    gin_gemm_kernel<<<ggrd, gblk, 0, stream>>>(x, agg, W1, b1, out);

    // Layer 2: agg2 = scatter(h1); out = (h1 + agg2) @ W2^T + b2 (in-place)
    gin_zero_kernel<<<zgrd, zblk, 0, stream>>>((float4*)agg, n4);
    gin_scatter_kernel<<<sgrd, sblk, 0, stream>>>(out, src, dst, agg);
    gin_gemm_kernel<<<ggrd, gblk, 0, stream>>>(out, agg, W2, b2, out);
}